// Binarize_66743791780423
// MI455X (gfx1250) — compile-verified
//
#include <hip/hip_runtime.h>
#include <hip/hip_bf16.h>
#include <stdint.h>

// Binarize: alpha[c] = sum(|x[c,:]|) / count(x[c,:] != 0); out = sign(x)*alpha[c]
// x: 4096 x 16384 fp32. Pure bandwidth kernel: 256MiB in + 256MiB out -> ~23us @ 23.3TB/s.
// Data path: one TDM tensor_load_to_lds per row (64KiB tile), compute from LDS/registers,
// single HBM read + single HBM write.

#define N_COLS   16384
#define BLOCK    1024                 // 32 wave32 per workgroup
#define V4_PER_T 4                    // 4 x float4 = 16 elements per thread

typedef __attribute__((ext_vector_type(4))) unsigned int u32x4;
typedef __attribute__((ext_vector_type(8))) int          i32x8;
typedef __attribute__((ext_vector_type(4))) int          i32x4;

// Build D# and issue one TDM load of a 16384-element fp32 row into LDS.
__device__ __forceinline__ void tdm_load_row(uint32_t lds_off, uint64_t gaddr) {
    // ---- D# group 0 (128b) ----
    u32x4 g0;
    g0.x = 1u;                                              // count=1 (valid), user mode
    g0.y = lds_off;                                         // lds_addr (byte offset)
    g0.z = (uint32_t)(gaddr & 0xFFFFFFFFu);                 // global_addr[31:0]
    g0.w = (uint32_t)((gaddr >> 32) & 0x1FFFFFFu)           // global_addr[56:32]
         | (2u << 30);                                      // type = 2 ("image")
    // ---- D# group 1 (256b) ----
    // data_size=2 (4B); tensor_dim0=16384; tensor_dim1=1; tile_dim0=16384; tile_dim1=1;
    // tensor_dim0_stride=16384 elements; no multicast, no pad, no iterate, no barrier.
    i32x8 g1 = {
        (int)(2u << 16),                                    // [17:16] data_size = 4B
        (int)(((uint32_t)N_COLS & 0xFFFFu) << 16),          // [31:16] tensor_dim0 lo16
        (int)(1u << 16),                                    // [15:0] dim0 hi16=0 | [31:16] tensor_dim1=1
        (int)(((uint32_t)N_COLS & 0xFFFFu) << 16),          // [31:16] tile_dim0 = 16384
        (int)1,                                             // [15:0] tile_dim1=1 | tile_dim2=0
        (int)N_COLS,                                        // tensor_dim0_stride lo32
        0,                                                  // stride hi16 | tensor_dim1_stride lo16
        0                                                   // tensor_dim1_stride hi32
    };
#if __has_builtin(__builtin_amdgcn_tensor_load_to_lds)
    // amdgpu-toolchain (clang-23) form: 6 args (u32x4, i32x8, i32x4, i32x4, i32x8, i32 cpol)
    i32x4 z4 = {0, 0, 0, 0};
    i32x8 z8 = {0, 0, 0, 0, 0, 0, 0, 0};
    __builtin_amdgcn_tensor_load_to_lds(g0, g1, z4, z4, z8, 0);
#else
    asm volatile("tensor_load_to_lds %0, %1" :: "s"(g0), "s"(g1) : "memory");
#endif
}

__global__ __launch_bounds__(BLOCK) void Binarize_66743791780423_kernel(
        const float* __restrict__ x, float* __restrict__ out) {
    __shared__ __align__(16) float row[N_COLS];   // 64 KiB tile (320 KiB LDS per WGP)
    __shared__ float red[2 * 32 + 1];             // per-wave partials + alpha slot

    const int t = (int)threadIdx.x;
    const uint64_t row_elem_base = (uint64_t)blockIdx.x * (uint64_t)N_COLS;

    // ---- Stage 1: TDM DMA the row into LDS (issued exactly once, by wave 0) ----
    // TDM ignores EXEC, so gate with a *scalar* branch: other waves jump around it.
    const int wave = __builtin_amdgcn_readfirstlane(t >> 5);
    if (wave == 0) {
        const uint32_t lds_off = (uint32_t)(uintptr_t)&row[0];
        const uint64_t gaddr   = (uint64_t)(uintptr_t)x + row_elem_base * sizeof(float);
        tdm_load_row(lds_off, gaddr);
#if __has_builtin(__builtin_amdgcn_s_wait_tensorcnt)
        __builtin_amdgcn_s_wait_tensorcnt(0);
#else
        asm volatile("s_wait_tensorcnt 0x0" ::: "memory");
#endif
    }
    __syncthreads();

    // ---- Stage 2: 4x ds_load_b128 per thread, keep payload in registers ----
    const float4* row4 = (const float4*)row;
    float4 v[V4_PER_T];
    float s = 0.0f, c = 0.0f;
#pragma unroll
    for (int i = 0; i < V4_PER_T; ++i) {
        v[i] = row4[t + i * BLOCK];               // stride-1024 float4s: coalesced, banks spread
        s += __builtin_fabsf(v[i].x) + __builtin_fabsf(v[i].y)
           + __builtin_fabsf(v[i].z) + __builtin_fabsf(v[i].w);
        c += (v[i].x != 0.0f ? 1.0f : 0.0f) + (v[i].y != 0.0f ? 1.0f : 0.0f)
           + (v[i].z != 0.0f ? 1.0f : 0.0f) + (v[i].w != 0.0f ? 1.0f : 0.0f);
    }

    // ---- Stage 3: wave32 shuffle reduce -> LDS -> wave0 reduce -> alpha ----
#pragma unroll
    for (int off = 16; off > 0; off >>= 1) {
        s += __shfl_down(s, off, 32);
        c += __shfl_down(c, off, 32);
    }
    if ((t & 31) == 0) {
        red[(t >> 5) * 2 + 0] = s;
        red[(t >> 5) * 2 + 1] = c;
    }
    __syncthreads();
    if (t < 32) {
        float s2 = red[t * 2 + 0];
        float c2 = red[t * 2 + 1];
#pragma unroll
        for (int off = 16; off > 0; off >>= 1) {
            s2 += __shfl_down(s2, off, 32);
            c2 += __shfl_down(c2, off, 32);
        }
        if (t == 0) red[64] = s2 / c2;            // alpha (count > 0 for normal data)
    }
    __syncthreads();
    const float alpha = red[64];

    // ---- Stage 4: write sign(x)*alpha straight from registers (coalesced b128) ----
    float4* out4 = (float4*)(out + row_elem_base);
#pragma unroll
    for (int i = 0; i < V4_PER_T; ++i) {
        float4 o;
        o.x = (v[i].x > 0.0f) ? alpha : -alpha;
        o.y = (v[i].y > 0.0f) ? alpha : -alpha;
        o.z = (v[i].z > 0.0f) ? alpha : -alpha;
        o.w = (v[i].w > 0.0f) ? alpha : -alpha;
        out4[t + i * BLOCK] = o;
    }
}

extern "C" void kernel_launch(void* const* d_in, const int* in_sizes, int n_in,
                              void* d_out, int out_size, void* d_ws, size_t ws_size,
                              hipStream_t stream) {
    const float* x = (const float*)d_in[0];
    float* out = (float*)d_out;
    const int total = in_sizes[0];                // 4096 * 16384
    const int n_rows = total / N_COLS;            // 4096 channels
    Binarize_66743791780423_kernel<<<n_rows, BLOCK, 0, stream>>>(x, out);
}